// CudaTensorProduct_19189913879101
// MI455X (gfx1250) — compile-verified
//
#include <hip/hip_runtime.h>

// Sparse Clebsch-Gordan tensor product for irreps 6x0e+4x1o+3x2e+2x3o (x2).
// B=16384, D1=D2=47, DOUT=2209.  Output-write bound (~145MB @ 23.3TB/s ~6.5us).
// Per-(l1,l2,l3) path contraction as f32 WMMA 16x16x4 tiles:
//   C[16 batch x 16 m3(pad)] += A[16 batch x 4 k] * B[4 k x 16 m3(pad)]
// B operands + k->(m1,m2) maps are pre-packed per triple in lane-pair order
// (one b64 load each per WMMA).  Inputs staged to LDS with async-to-LDS b128;
// outputs accumulated in a contiguous LDS tile and flushed with
// store-async-from-LDS b128 (fully coalesced).  Tables rebuilt on device each
// call (graph-capture safe, deterministic).

#define D1   47
#define D2   47
#define DOUT 2209
#define NIRR 15
#define NPATH_MAX 445
#define KMAP_TOTAL 1184      // sum over 44 triples of Kp = 4*ceil(n1*n2/4)
#define WPAD_TOTAL 18944     // sum of Kp*16
#define BT   16
#define WAVES 8

typedef __attribute__((ext_vector_type(2))) float v2f;
typedef __attribute__((ext_vector_type(8))) float v8f;
typedef __attribute__((ext_vector_type(2))) int   v2i;
typedef __attribute__((ext_vector_type(4))) int   v4i;

#if defined(__has_builtin)
#if __has_builtin(__builtin_amdgcn_wmma_f32_16x16x4_f32)
#define USE_WMMA 1
#endif
#if __has_builtin(__builtin_amdgcn_global_load_async_to_lds_b128) && \
    __has_builtin(__builtin_amdgcn_s_wait_asynccnt)
#define USE_ASYNC 1
#endif
#if __has_builtin(__builtin_amdgcn_global_store_async_from_lds_b128)
#define USE_AST 1
#endif
#endif
#ifndef USE_WMMA
#define USE_WMMA 0
#endif
#ifndef USE_ASYNC
#define USE_ASYNC 0
#endif
#ifndef USE_AST
#define USE_AST 0
#endif

struct PathMeta { int o1, o2, out_off, n3, Kp, koff, wpoff, pad; };

// Packed pair layouts (per triple):
//   kmap2[(koff>>1) + c*2 + h]        = { map[4c+2h], map[4c+2h+1] }
//   wpad2[(wpoff>>1) + c*32 + h*16+n] = { W[4c+2h][n], W[4c+2h+1][n] }
__device__ v2f      g_wpad2[WPAD_TOTAL/2];
__device__ v2i      g_kmap2[KMAP_TOTAL/2];
__device__ PathMeta g_meta[NPATH_MAX];
__device__ int      g_npath;
__device__ int      g_bucket[7][128][4];

__device__ __forceinline__ int iabs_(int x){ return x < 0 ? -x : x; }
__device__ __forceinline__ int imax_(int a, int b){ return a > b ? a : b; }
__device__ __forceinline__ int imin_(int a, int b){ return a < b ? a : b; }

// Padded-table offsets (element units) for triple (l1,l2,l3); enumeration
// order: l1:0..3, l2:0..3, l3:|l1-l2|..l1+l2  (44 triples).
__device__ void triOffsets(int l1, int l2, int l3, int* koff, int* wpoff){
  int ko = 0, wo = 0;
  for(int a = 0; a < 4; ++a)
    for(int b = 0; b < 4; ++b){
      int K = (2*a+1)*(2*b+1);
      int Kp = (K + 3) & ~3;
      for(int c = iabs_(a-b); c <= a+b; ++c){
        if(a == l1 && b == l2 && c == l3){ *koff = ko; *wpoff = wo; return; }
        ko += Kp; wo += Kp*16;
      }
    }
  *koff = 0; *wpoff = 0;
}

struct cplx { double re, im; };
__device__ __forceinline__ cplx cmul(cplx a, cplx b){
  cplx r; r.re = a.re*b.re - a.im*b.im; r.im = a.re*b.im + a.im*b.re; return r;
}

__device__ double dfact(int n){ double r = 1.0; for(int i = 2; i <= n; ++i) r *= (double)i; return r; }

// Racah formula <j1 m1 j2 m2 | j3 m3> (doubles exact enough: args <= 13!)
__device__ double su2_cg_coeff(int j1,int m1,int j2,int m2,int j3,int m3){
  if(m3 != m1 + m2) return 0.0;
  int vmin = imax_(imax_(-j1 + j2 + m3, -j1 + m1), 0);
  int vmax = imin_(imin_(j2 + j3 + m1, j3 - j1 + j2), j3 + m3);
  double C = sqrt((double)(2*j3+1) * dfact(j3+j1-j2) * dfact(j3-j1+j2) * dfact(j1+j2-j3)
                  * dfact(j3+m3) * dfact(j3-m3)
                  / (dfact(j1+j2+j3+1) * dfact(j1-m1) * dfact(j1+m1) * dfact(j2-m2) * dfact(j2+m2)));
  double S = 0.0;
  for(int v = vmin; v <= vmax; ++v){
    double sgn = ((v + j2 + m2) & 1) ? -1.0 : 1.0;
    S += sgn * dfact(j2+j3+m1-v) * dfact(j1-m1+v)
         / (dfact(v) * dfact(j3-j1+j2-v) * dfact(j3+m3-v) * dfact(v+j1-j2-m3));
  }
  return C * S;
}

// e3nn change-of-basis real SH -> complex SH, including (-i)^l phase.
__device__ void build_q(int l, cplx q[7][7]){
  for(int i = 0; i < 7; ++i) for(int j = 0; j < 7; ++j){ q[i][j].re = 0.0; q[i][j].im = 0.0; }
  const double is2 = 0.70710678118654752440;
  for(int m = -l; m < 0; ++m){
    q[l+m][l-m].re = is2;     // column l+|m|
    q[l+m][l+m].im = -is2;    // column l-|m|
  }
  q[l][l].re = 1.0;
  for(int m = 1; m <= l; ++m){
    double s = (m & 1) ? -1.0 : 1.0;
    q[l+m][l+m].re = s * is2;
    q[l+m][l-m].im = s * is2;
  }
  cplx ph; int lm = l & 3;
  if(lm == 0){ ph.re = 1; ph.im = 0; } else if(lm == 1){ ph.re = 0; ph.im = -1; }
  else if(lm == 2){ ph.re = -1; ph.im = 0; } else { ph.re = 0; ph.im = 1; }
  for(int i = 0; i < 7; ++i) for(int j = 0; j < 7; ++j) q[i][j] = cmul(q[i][j], ph);
}

// One thread per distinct (l1,l2,l3) triple: dense real w3j * sqrt(2*l3+1),
// emitted directly in WMMA-ready pair-packed layout + k->(m1,m2) map.
__global__ void build_w3j_kernel(){
  const int t = threadIdx.x;
  int l1 = -1, l2 = -1, l3 = -1;
  {
    int idx = 0; bool found = false;
    for(int a = 0; a < 4 && !found; ++a)
      for(int b = 0; b < 4 && !found; ++b)
        for(int c = iabs_(a-b); c <= a+b; ++c){
          if(idx == t){ l1 = a; l2 = b; l3 = c; found = true; break; }
          idx++;
        }
    if(!found) return;
  }
  const int n1 = 2*l1+1, n2 = 2*l2+1, n3 = 2*l3+1;

  double C[7][7][7];
  for(int i = 0; i < 7; ++i) for(int k = 0; k < 7; ++k) for(int n = 0; n < 7; ++n) C[i][k][n] = 0.0;
  for(int m1 = -l1; m1 <= l1; ++m1)
    for(int m2 = -l2; m2 <= l2; ++m2){
      int m3 = m1 + m2;
      if(iabs_(m3) <= l3) C[l1+m1][l2+m2][l3+m3] = su2_cg_coeff(l1, m1, l2, m2, l3, m3);
    }
  // (reference divides by sqrt(2j3+1) here; it cancels in the normalization)

  cplx Q1[7][7], Q2[7][7], Q3[7][7];
  build_q(l1, Q1); build_q(l2, Q2); build_q(l3, Q3);

  // einsum('ij,kl,mn,ikn->jlm', Q1, Q2, conj(Q3.T), C); keep real part.
  double w[343]; double nrm = 0.0;
  for(int j = 0; j < n1; ++j)
    for(int l = 0; l < n2; ++l)
      for(int m = 0; m < n3; ++m){
        double ar = 0.0;
        for(int i = 0; i < n1; ++i)
          for(int k = 0; k < n2; ++k){
            int n = (i - l1) + (k - l2) + l3;       // CG nonzero only here
            if(n < 0 || n >= n3) continue;
            double cv = C[i][k][n];
            if(cv == 0.0) continue;
            cplx t1 = cmul(Q1[i][j], Q2[k][l]);
            cplx q3 = Q3[n][m]; q3.im = -q3.im;     // conj(Q3[n][m])
            cplx t2 = cmul(t1, q3);
            ar += t2.re * cv;
          }
        w[(j*n2 + l)*n3 + m] = ar; nrm += ar * ar;
      }
  double scale = sqrt((double)n3) / sqrt(nrm);      // unit-norm, * sqrt(2*l3+1)

  // Emit pair-packed B operand and k-map for this triple.
  const int K  = n1 * n2;
  const int Kp = (K + 3) & ~3;
  int koff, wpoff; triOffsets(l1, l2, l3, &koff, &wpoff);
  const int k2 = koff >> 1, w2 = wpoff >> 1;
  for(int cc = 0; cc < (Kp >> 2); ++cc){
    for(int h = 0; h < 2; ++h){
      int r0 = cc*4 + h*2;
      int mres[2];
      float wv0[16], wv1[16];
      for(int q = 0; q < 2; ++q){
        int k = r0 + q;
        int kk = (k < K) ? k : 0;            // sentinel: pad rows map to valid
        int m2 = kk / n1, m1 = kk - m2*n1;   // indices; B row is zero there
        mres[q] = m1 | (m2 << 8);
        for(int nn = 0; nn < 16; ++nn){
          float v = 0.0f;
          if(k < K && nn < n3) v = (float)(w[(m1*n2 + m2)*n3 + nn] * scale);
          if(q == 0) wv0[nn] = v; else wv1[nn] = v;
        }
      }
      v2i mm; mm.x = mres[0]; mm.y = mres[1];
      g_kmap2[k2 + cc*2 + h] = mm;
      for(int nn = 0; nn < 16; ++nn){
        v2f wv; wv.x = wv0[nn]; wv.y = wv1[nn];
        g_wpad2[w2 + cc*32 + h*16 + nn] = wv;
      }
    }
  }
}

// Single thread: exact replication of _build_layout path ordering.
__global__ void build_layout_kernel(){
  if(threadIdx.x != 0 || blockIdx.x != 0) return;
  const int ls[NIRR] = {0,0,0,0,0,0, 1,1,1,1, 2,2,2, 3,3};
  int bcnt[7]; for(int i = 0; i < 7; ++i) bcnt[i] = 0;
  int o1 = 0;
  for(int i1 = 0; i1 < NIRR; ++i1){
    int l1 = ls[i1];
    int o2 = 0;
    for(int i2 = 0; i2 < NIRR; ++i2){
      int l2 = ls[i2];
      for(int l3 = iabs_(l1-l2); l3 <= l1+l2; ++l3){
        int c = bcnt[l3]++;
        g_bucket[l3][c][0] = l1; g_bucket[l3][c][1] = l2;
        g_bucket[l3][c][2] = o1; g_bucket[l3][c][3] = o2;
      }
      o2 += 2*l2 + 1;
    }
    o1 += 2*l1 + 1;
  }
  int out_off = 0, pc = 0;
  for(int l3 = 0; l3 < 7; ++l3){
    int nb = bcnt[l3];
    // stable insertion sort by key l1*LMAX2 + l2  (LMAX2 == 3; key collisions
    // keep insertion order, matching Python's stable list.sort)
    for(int i = 1; i < nb; ++i){
      int e0 = g_bucket[l3][i][0], e1 = g_bucket[l3][i][1];
      int e2 = g_bucket[l3][i][2], e3 = g_bucket[l3][i][3];
      int key = e0*3 + e1;
      int j = i - 1;
      while(j >= 0 && (g_bucket[l3][j][0]*3 + g_bucket[l3][j][1]) > key){
        g_bucket[l3][j+1][0] = g_bucket[l3][j][0]; g_bucket[l3][j+1][1] = g_bucket[l3][j][1];
        g_bucket[l3][j+1][2] = g_bucket[l3][j][2]; g_bucket[l3][j+1][3] = g_bucket[l3][j][3];
        --j;
      }
      g_bucket[l3][j+1][0] = e0; g_bucket[l3][j+1][1] = e1;
      g_bucket[l3][j+1][2] = e2; g_bucket[l3][j+1][3] = e3;
    }
    for(int i = 0; i < nb; ++i){
      int l1 = g_bucket[l3][i][0], l2 = g_bucket[l3][i][1];
      int K = (2*l1+1)*(2*l2+1);
      PathMeta m;
      m.o1 = g_bucket[l3][i][2]; m.o2 = g_bucket[l3][i][3];
      m.out_off = out_off; m.n3 = 2*l3+1;
      m.Kp = (K + 3) & ~3;
      triOffsets(l1, l2, l3, &m.koff, &m.wpoff);
      m.pad = 0;
      g_meta[pc++] = m;
      out_off += 2*l3 + 1;
    }
  }
  g_npath = pc;   // == 445, out_off == 2209
}

__global__ __launch_bounds__(256) void tp_kernel(const float* __restrict__ in1,
                                                 const float* __restrict__ in2,
                                                 float* __restrict__ out,
                                                 int batch){
  typedef __attribute__((address_space(1))) v4i* gp4_t;
  typedef __attribute__((address_space(3))) v4i* lp4_t;

  // Flat input tiles: 16 rows x 47 floats, contiguous (stride-47 lane access
  // is bank-conflict-free, gcd(47,64)==1), 16B aligned for async b128.
  __shared__ __align__(16) float s1f[BT * D1];
  __shared__ __align__(16) float s2f[BT * D2];
#if USE_AST
  // Contiguous output tile (16 x 2209 = 141376B; CDNA5 allows 320KB/WG LDS).
  __shared__ __align__(16) float so[BT * DOUT];
#endif

  const int tid  = threadIdx.x;
  const int b0   = blockIdx.x * BT;
  const bool full = (b0 + BT <= batch);

#if USE_ASYNC
  if(full){
    // 3008B tile == 188 x b128; base offset b0*47*4 = blockIdx*3008 (16B mult)
    if(tid < BT*D1/4){
      __builtin_amdgcn_global_load_async_to_lds_b128(
          (gp4_t)(in1 + (size_t)b0*D1 + tid*4), (lp4_t)&s1f[tid*4], 0, 0);
      __builtin_amdgcn_global_load_async_to_lds_b128(
          (gp4_t)(in2 + (size_t)b0*D2 + tid*4), (lp4_t)&s2f[tid*4], 0, 0);
    }
    __builtin_amdgcn_s_wait_asynccnt(0);
  } else
#endif
  {
    for(int i = tid; i < BT * D1; i += 256){
      int b = b0 + i / D1;
      s1f[i] = (b < batch) ? in1[(size_t)b0 * D1 + i] : 0.0f;
      s2f[i] = (b < batch) ? in2[(size_t)b0 * D2 + i] : 0.0f;
    }
  }
  __syncthreads();

  const int lane = tid & 31;
  const int wave = tid >> 5;
  const int bl   = lane & 15;        // A-operand batch row for this lane
  const int hi   = lane >> 4;        // 0: low half, 1: high half
  const int n    = lane & 15;        // B/C column (m3)
  const int b47  = bl * D1;
  const int npath = g_npath;

  for(int p = wave; p < npath; p += WAVES){
    PathMeta pm = g_meta[p];
    __builtin_prefetch(&g_wpad2[pm.wpoff >> 1], 0, 1);   // global_prefetch_b8
    const v2i* km2 = g_kmap2 + (pm.koff >> 1) + hi;      // + c*2
    const v2f* wp2 = g_wpad2 + (pm.wpoff >> 1) + hi*16 + n;  // + c*32
    const float* s1r = &s1f[b47 + pm.o1];
    const float* s2r = &s2f[b47 + pm.o2];
    const int nch = pm.Kp >> 2;

    v8f c; for(int r = 0; r < 8; ++r) c[r] = 0.0f;
#if USE_WMMA
    for(int cc = 0; cc < nch; ++cc){
      v2i ix = km2[cc*2];          // one b64: k-map pair
      v2f b  = wp2[cc*32];         // one b64: B-operand pair (VGPR0,VGPR1 rows)
      v2f a;
      a.x = s1r[ix.x & 0xff] * s2r[ix.x >> 8];
      a.y = s1r[ix.y & 0xff] * s2r[ix.y >> 8];
      // D = A(16x4) * B(4x16) + C   (f32, RNE)
      c = __builtin_amdgcn_wmma_f32_16x16x4_f32(false, a, false, b,
                                                (short)0, c, false, false);
    }
#else
    for(int r = 0; r < 8; ++r){
      int M47 = (r + hi*8) * D1;
      float acc = 0.0f;
      for(int cc = 0; cc < nch; ++cc)
        for(int h = 0; h < 2; ++h){
          v2i ix = g_kmap2[(pm.koff >> 1) + cc*2 + h];
          v2f wv = g_wpad2[(pm.wpoff >> 1) + cc*32 + h*16 + n];
          acc += s1f[M47 + pm.o1 + (ix.x & 0xff)] * s2f[M47 + pm.o2 + (ix.x >> 8)] * wv.x;
          acc += s1f[M47 + pm.o1 + (ix.y & 0xff)] * s2f[M47 + pm.o2 + (ix.y >> 8)] * wv.y;
        }
      c[r] = acc;
    }
#endif

#if USE_AST
    if(n < pm.n3){
      // rows hi*8..hi*8+7; lanes n=0..n3-1 hit consecutive banks; the two
      // half-waves differ by 8 banks (8*2209 mod 64 == 8) -> conflict-free.
      #pragma unroll
      for(int r = 0; r < 8; ++r)
        so[(r + hi*8) * DOUT + pm.out_off + n] = c[r];
    }
#else
    if(n < pm.n3){
      float* orow = out + (size_t)(b0 + hi*8) * DOUT + pm.out_off + n;
      if(full){
        #pragma unroll
        for(int r = 0; r < 8; ++r) orow[(size_t)r * DOUT] = c[r];
      } else {
        for(int r = 0; r < 8; ++r)
          if(b0 + hi*8 + r < batch) orow[(size_t)r * DOUT] = c[r];
      }
    }
#endif
  }

#if USE_AST
  __syncthreads();
  if(full){
    // Flush the contiguous 141376B tile with coalesced async b128 stores.
    float* obase = out + (size_t)b0 * DOUT;
    for(int i = tid; i < (BT*DOUT)/4; i += 256)
      __builtin_amdgcn_global_store_async_from_lds_b128(
          (gp4_t)(obase + (size_t)i*4), (lp4_t)&so[i*4], 0, 0);
    // implicit s_wait_idle at s_endpgm covers outstanding ASYNCcnt
  } else {
    for(int i = tid; i < BT*DOUT; i += 256){
      int r = i / DOUT, col = i - r*DOUT;
      if(b0 + r < batch) out[(size_t)(b0 + r) * DOUT + col] = so[i];
    }
  }
#endif
}

extern "C" void kernel_launch(void* const* d_in, const int* in_sizes, int n_in,
                              void* d_out, int out_size, void* d_ws, size_t ws_size,
                              hipStream_t stream){
  (void)n_in; (void)out_size; (void)d_ws; (void)ws_size;
  const float* in1 = (const float*)d_in[0];
  const float* in2 = (const float*)d_in[1];
  float* out = (float*)d_out;
  const int batch = in_sizes[0] / D1;

  // Rebuild constant tables on-device every call (deterministic, capture-safe).
  build_w3j_kernel<<<1, 64, 0, stream>>>();
  build_layout_kernel<<<1, 1, 0, stream>>>();

  const int nblocks = (batch + BT - 1) / BT;
  tp_kernel<<<nblocks, 256, 0, stream>>>(in1, in2, out, batch);
}